// ProposalLayer_2388001816805
// MI455X (gfx1250) — compile-verified
//
#include <hip/hip_runtime.h>
#include <stdint.h>
#include <math.h>

// ---------------------------------------------------------------------------
// ProposalLayer for MI455X (gfx1250, wave32)
//   B=8 batches, N=262144 anchors, top-6000 pre-NMS, 1000 proposals, IoU>0.7
//
// Latency-bound: the 1000-step greedy NMS is a serial chain. One 1024-thread
// workgroup per batch; NMS state lives in registers (boxes/areas/valid bits),
// pick selection via wave32 __shfl_xor min-reduce; box tile DMA'd global->LDS
// with the CDNA5 Tensor Data Mover. Top-k via 2-level radix histogram with a
// parallel suffix scan + 8192-wide bitonic sort in LDS.
// ---------------------------------------------------------------------------

#define BATCH       8
#define N_ANCH      262144
#define PRE_NMS     6000
#define PROPOSALS   1000
#define NMS_T       0.7f
#define CAND_CAP    8192                   // power of two >= 6000 + radix slack
#define SLOTS       6                      // ceil(6000 / 1024) boxes per thread

typedef unsigned int  tdm_u32x4 __attribute__((ext_vector_type(4)));
typedef int           tdm_i32x8 __attribute__((ext_vector_type(8)));
typedef int           tdm_i32x4 __attribute__((ext_vector_type(4)));

// ---------------------------------------------------------------------------
// Parallel suffix-scan over bufA[0..4095] (ping-pong with bufB), then find the
// unique crossing bin b with S[b] >= K > S[b+1]. Returns via shared scalars.
// Must be called by all 1024 threads.
// ---------------------------------------------------------------------------
__device__ __forceinline__ void suffix_select(
    unsigned int* bufA, unsigned int* bufB, unsigned int K,
    unsigned int* sh_bin, unsigned int* sh_above, int tid)
{
    unsigned int* src = bufA;
    unsigned int* dst = bufB;
    for (int off = 1; off < 4096; off <<= 1) {
        for (int i = tid; i < 4096; i += 1024)
            dst[i] = src[i] + ((i + off < 4096) ? src[i + off] : 0u);
        __syncthreads();
        unsigned int* t = src; src = dst; dst = t;
    }
    // result (inclusive suffix sums, non-increasing) now in src
    for (int i = tid; i < 4096; i += 1024) {
        unsigned int s = src[i];
        unsigned int snext = (i < 4095) ? src[i + 1] : 0u;
        if (s >= K && snext < K) { *sh_bin = (unsigned int)i; *sh_above = snext; }
    }
    __syncthreads();
}

// ===========================================================================
// Kernel 1: per-batch exact top-6000 (radix-select + bitonic sort) and box
// decode (deltas * std, exp, clip). Writes sorted boxes to d_ws.
// ===========================================================================
__global__ __launch_bounds__(1024) void topk_decode_kernel(
    const float* __restrict__ probs,    // (B, N, 2)
    const float* __restrict__ bbox,     // (B, N, 4)
    const float* __restrict__ anchors,  // (B, N, 4)
    float* __restrict__ boxesWS)        // (B, 6000, 4)
{
    extern __shared__ unsigned char dynsmem[];
    unsigned long long* cand = (unsigned long long*)dynsmem;  // 8192 * 8B = 64KB
    unsigned int* bufA = (unsigned int*)dynsmem;              // 4096 bins (aliased)
    unsigned int* bufB = bufA + 4096;                         // scan ping-pong

    __shared__ unsigned int sh_b1, sh_above, sh_b2, sh_above2, sh_cnt;

    const int b   = blockIdx.x;
    const int tid = threadIdx.x;
    const float* sc = probs + (size_t)b * N_ANCH * 2;

    // ---- level-1 histogram over key[31:20] (scores >= 0: uint order == float order)
    for (int i = tid; i < 4096; i += 1024) bufA[i] = 0u;
    __syncthreads();
    for (int i = tid; i < N_ANCH; i += 1024) {
        unsigned int key = __float_as_uint(sc[2 * i + 1]);
        atomicAdd(&bufA[key >> 20], 1u);
    }
    __syncthreads();
    suffix_select(bufA, bufB, PRE_NMS, &sh_b1, &sh_above, tid);
    const unsigned int b1   = sh_b1;
    const unsigned int need = PRE_NMS - sh_above;   // >= 1

    // ---- level-2 histogram over key[19:8] within bin b1
    for (int i = tid; i < 4096; i += 1024) bufA[i] = 0u;
    __syncthreads();
    for (int i = tid; i < N_ANCH; i += 1024) {
        unsigned int key = __float_as_uint(sc[2 * i + 1]);
        if ((key >> 20) == b1) atomicAdd(&bufA[(key >> 8) & 0xFFFu], 1u);
    }
    __syncthreads();
    suffix_select(bufA, bufB, need, &sh_b2, &sh_above2, tid);
    const unsigned int T24 = (b1 << 12) | sh_b2;    // 24-bit prefix threshold
    if (tid == 0) sh_cnt = 0u;
    __syncthreads();

    // ---- collect candidates with key>>8 >= T24 (aliases histograms; done)
    for (int i = tid; i < N_ANCH; i += 1024) {
        unsigned int key = __float_as_uint(sc[2 * i + 1]);
        if ((key >> 8) >= T24) {
            unsigned int pos = atomicAdd(&sh_cnt, 1u);
            if (pos < CAND_CAP)
                cand[pos] = ((unsigned long long)key << 32) |
                            (unsigned long long)(~(unsigned int)i);
        }
    }
    __syncthreads();
    unsigned int cnt = sh_cnt; if (cnt > CAND_CAP) cnt = CAND_CAP;
    for (unsigned int i = cnt + tid; i < CAND_CAP; i += 1024) cand[i] = 0ull;
    __syncthreads();

    // ---- bitonic sort, descending: (score desc, index asc) == lax.top_k order
    for (unsigned int k = 2; k <= CAND_CAP; k <<= 1) {
        for (unsigned int j = k >> 1; j > 0; j >>= 1) {
            for (unsigned int t = tid; t < (CAND_CAP / 2); t += 1024) {
                unsigned int i = ((t / j) * (j << 1)) + (t & (j - 1));
                unsigned int p = i + j;
                unsigned long long a = cand[i], c = cand[p];
                bool desc = ((i & k) == 0u);
                if (desc ? (a < c) : (a > c)) { cand[i] = c; cand[p] = a; }
            }
            __syncthreads();
        }
    }

    // ---- decode top-6000 boxes: deltas*std, exp, clip; stage to d_ws
    const float* dl = bbox    + (size_t)b * N_ANCH * 4;
    const float* an = anchors + (size_t)b * N_ANCH * 4;
    float* outB = boxesWS + (size_t)b * PRE_NMS * 4;
    for (int t = tid; t < PRE_NMS; t += 1024) {
        unsigned int idx = ~(unsigned int)(cand[t] & 0xFFFFFFFFull);
        float a0 = an[4 * idx + 0], a1 = an[4 * idx + 1];
        float a2 = an[4 * idx + 2], a3 = an[4 * idx + 3];
        float d0 = dl[4 * idx + 0] * 0.1f, d1 = dl[4 * idx + 1] * 0.1f;
        float d2 = dl[4 * idx + 2] * 0.2f, d3 = dl[4 * idx + 3] * 0.2f;
        float h = a2 - a0, w = a3 - a1;
        float cy = a0 + 0.5f * h + d0 * h;
        float cx = a1 + 0.5f * w + d1 * w;
        h *= expf(d2);
        w *= expf(d3);
        float y1 = cy - 0.5f * h, x1 = cx - 0.5f * w;
        float y2 = cy + 0.5f * h, x2 = cx + 0.5f * w;
        outB[4 * t + 0] = fminf(fmaxf(y1, 0.0f), 1.0f);
        outB[4 * t + 1] = fminf(fmaxf(x1, 0.0f), 1.0f);
        outB[4 * t + 2] = fminf(fmaxf(y2, 0.0f), 1.0f);
        outB[4 * t + 3] = fminf(fmaxf(x2, 0.0f), 1.0f);
    }
}

// ===========================================================================
// Kernel 2: per-batch sequential greedy NMS, state fully register-resident.
// Box tile loaded global->LDS via the Tensor Data Mover, then cached into
// registers (each thread owns boxes tid + 1024*k, k = 0..5).
// ===========================================================================
__global__ __launch_bounds__(1024) void nms_kernel(
    const float* __restrict__ boxesWS,  // (B, 6000, 4)
    float* __restrict__ out)            // (B, 1000, 4)
{
    extern __shared__ unsigned char dynsmem[];
    float* bx = (float*)dynsmem;                       // 6000*4 floats = 96KB

    __shared__ float sh_pb[5];                         // picked box + area
    __shared__ int   sh_pick;
    __shared__ int   sh_wmin[32];

    const int b   = blockIdx.x;
    const int tid = threadIdx.x;
    const float* src = boxesWS + (size_t)b * PRE_NMS * 4;

#if __has_builtin(__builtin_amdgcn_tensor_load_to_lds)
    // TDM: 1-D tile of 24000 dwords, data_size=4B, global -> LDS.
    if (tid < 32) {   // one wave issues the DMA
        unsigned long long ga = (unsigned long long)(uintptr_t)src;
        unsigned int la = (unsigned int)(uintptr_t)bx;   // flat low 32 = LDS byte offset
        const unsigned int LEN = PRE_NMS * 4;            // 24000 dwords

        tdm_u32x4 g0;
        g0[0] = 1u;                                      // count=1, load, no gather
        g0[1] = la;                                      // lds_addr
        g0[2] = (unsigned int)ga;                        // global_addr[31:0]
        g0[3] = (unsigned int)((ga >> 32) & 0x1FFFFFFu)  // global_addr[56:32]
              | (2u << 30);                              // type = 2 ("image")

        tdm_i32x8 g1;
        g1[0] = (int)(2u << 16);                         // wg_mask=0, data_size=2 (4B)
        g1[1] = (int)((LEN & 0xFFFFu) << 16);            // tensor_dim0[15:0]
        g1[2] = (int)(((LEN >> 16) & 0xFFFFu) | (1u << 16)); // dim0 hi | tensor_dim1=1
        g1[3] = (int)((LEN & 0xFFFFu) << 16);            // tile_dim0 = 24000
        g1[4] = 1;                                       // tile_dim1=1, tile_dim2=0
        g1[5] = (int)LEN;                                // tensor_dim0_stride[31:0]
        g1[6] = (int)((LEN & 0xFFFFu) << 16);            // stride0 hi=0 | stride1 lo
        g1[7] = 0;                                       // stride1 hi

        tdm_i32x4 g2 = {0, 0, 0, 0};
        tdm_i32x4 g3 = {0, 0, 0, 0};
        tdm_i32x8 g4 = {0, 0, 0, 0, 0, 0, 0, 0};
        __builtin_amdgcn_tensor_load_to_lds(g0, g1, g2, g3, g4, 0);
        __builtin_amdgcn_s_wait_tensorcnt(0);
    }
#else
    for (int i = tid; i < PRE_NMS * 4; i += 1024) bx[i] = src[i];
#endif
    __syncthreads();

    // ---- register cache: this thread owns boxes tid + 1024*k
    float rb[SLOTS][4];
    float ra[SLOTS];
    unsigned int mask = 0u;
#pragma unroll
    for (int k = 0; k < SLOTS; ++k) {
        int t = tid + (k << 10);
        if (t < PRE_NMS) {
            rb[k][0] = bx[4 * t + 0];
            rb[k][1] = bx[4 * t + 1];
            rb[k][2] = bx[4 * t + 2];
            rb[k][3] = bx[4 * t + 3];
            ra[k] = (rb[k][2] - rb[k][0]) * (rb[k][3] - rb[k][1]);
            mask |= (1u << k);
        } else {
            rb[k][0] = rb[k][1] = rb[k][2] = rb[k][3] = 0.0f;
            ra[k] = 0.0f;
        }
    }

    float* dst = out + (size_t)b * PROPOSALS * 4;
    const int INVALID = 0x7FFFFFFF;

    for (int it = 0; it < PROPOSALS; ++it) {
        // ---- min valid global index: smallest set k gives tid + 1024*k
        int local = INVALID;
#pragma unroll
        for (int k = SLOTS - 1; k >= 0; --k)
            if (mask & (1u << k)) local = tid + (k << 10);

        // wave32 min-reduce, then cross-wave reduce (two shuffle trees)
        int m = local;
#pragma unroll
        for (int off = 16; off >= 1; off >>= 1)
            m = min(m, __shfl_xor(m, off, 32));
        if ((tid & 31) == 0) sh_wmin[tid >> 5] = m;
        __syncthreads();
        if (tid < 32) {
            int v = sh_wmin[tid];
#pragma unroll
            for (int off = 16; off >= 1; off >>= 1)
                v = min(v, __shfl_xor(v, off, 32));
            if (tid == 0) sh_pick = v;
        }
        __syncthreads();
        const int pick = sh_pick;

        if (pick == INVALID) {   // no valid boxes remain -> zeros (matches ref)
            if (tid == 0) {
                dst[4 * it + 0] = 0.0f; dst[4 * it + 1] = 0.0f;
                dst[4 * it + 2] = 0.0f; dst[4 * it + 3] = 0.0f;
            }
            continue;
        }

        // ---- owner lane publishes the picked box + its area
        if (tid == (pick & 1023)) {
            const int k = pick >> 10;
            sh_pb[0] = rb[k][0]; sh_pb[1] = rb[k][1];
            sh_pb[2] = rb[k][2]; sh_pb[3] = rb[k][3];
            sh_pb[4] = ra[k];
        }
        __syncthreads();
        const float p0 = sh_pb[0], p1 = sh_pb[1], p2 = sh_pb[2], p3 = sh_pb[3];
        const float pa = sh_pb[4];
        if (tid == 0) {
            dst[4 * it + 0] = p0; dst[4 * it + 1] = p1;
            dst[4 * it + 2] = p2; dst[4 * it + 3] = p3;
        }

        // ---- suppress IoU > 0.7 (pure register VALU; self-IoU==1 clears pick,
        // unless area==0 -> iou==0 re-picks forever, matching the reference)
#pragma unroll
        for (int k = 0; k < SLOTS; ++k) {
            if (mask & (1u << k)) {
                float yy1 = fmaxf(p0, rb[k][0]);
                float xx1 = fmaxf(p1, rb[k][1]);
                float yy2 = fminf(p2, rb[k][2]);
                float xx2 = fminf(p3, rb[k][3]);
                float inter = fmaxf(yy2 - yy1, 0.0f) * fmaxf(xx2 - xx1, 0.0f);
                float iou = inter / (pa + ra[k] - inter + 1e-8f);
                if (iou > NMS_T) mask &= ~(1u << k);
            }
        }
    }
}

// ===========================================================================
extern "C" void kernel_launch(void* const* d_in, const int* in_sizes, int n_in,
                              void* d_out, int out_size, void* d_ws, size_t ws_size,
                              hipStream_t stream) {
    const float* probs   = (const float*)d_in[0];   // (8, 262144, 2)
    const float* bbox    = (const float*)d_in[1];   // (8, 262144, 4)
    const float* anchors = (const float*)d_in[2];   // (8, 262144, 4)
    float* boxesWS = (float*)d_ws;                  // needs 8*6000*4*4 = 768 KB

    const size_t lds1 = (size_t)CAND_CAP * sizeof(unsigned long long);   // 64 KB
    const size_t lds2 = (size_t)PRE_NMS * 4 * sizeof(float);             // 96 KB

    topk_decode_kernel<<<BATCH, 1024, lds1, stream>>>(probs, bbox, anchors, boxesWS);
    nms_kernel<<<BATCH, 1024, lds2, stream>>>(boxesWS, (float*)d_out);
}